// PennyLanePQC_30279519437348
// MI455X (gfx1250) — compile-verified
//
#include <hip/hip_runtime.h>

#define NTHR   256
#define NWAVES (NTHR / 32)
#define PI_F   3.14159265358979323846f

typedef float v2f __attribute__((ext_vector_type(2)));
typedef float v8f __attribute__((ext_vector_type(8)));

// ---------------------------------------------------------------------------
// conv-block 4x4 unitary construction (local index m: bit1 = qubit a, bit0 = b)
// ---------------------------------------------------------------------------
__device__ __forceinline__ void rz_local(float* re, float* im, int qbit, float t) {
  float c = __cosf(0.5f * t), s = __sinf(0.5f * t);
#pragma unroll
  for (int m = 0; m < 4; ++m) {
    float sg = ((m >> qbit) & 1) ? s : -s;       // e^{-it/2} for bit0, e^{+it/2} for bit1
    float r = re[m], i = im[m];
    re[m] = r * c - i * sg;
    im[m] = r * sg + i * c;
  }
}
__device__ __forceinline__ void ry_b_local(float* re, float* im, float t) {
  float c = __cosf(0.5f * t), s = __sinf(0.5f * t);
#pragma unroll
  for (int m = 0; m < 4; m += 2) {               // pairs (0,1),(2,3): qubit b = bit0
    float r0 = re[m], i0 = im[m], r1 = re[m + 1], i1 = im[m + 1];
    re[m]     = c * r0 - s * r1;  im[m]     = c * i0 - s * i1;
    re[m + 1] = s * r0 + c * r1;  im[m + 1] = s * i0 + c * i1;
  }
}
__device__ __forceinline__ void swap_local(float* re, float* im, int x, int y) {
  float t;
  t = re[x]; re[x] = re[y]; re[y] = t;
  t = im[x]; im[x] = im[y]; im[y] = t;
}

// threads 0..3 each build one column of U (exact _conv gate order)
__device__ void build_conv_U(const float* p, float2* U, int tid) {
  if (tid < 4) {
    float re[4] = {0.f, 0.f, 0.f, 0.f}, im[4] = {0.f, 0.f, 0.f, 0.f};
    re[tid] = 1.f;
    rz_local(re, im, 0, -0.5f * PI_F);  // rz(-pi/2) on b
    swap_local(re, im, 1, 3);           // CNOT ctrl=b tgt=a
    rz_local(re, im, 1, p[0]);          // rz(p0) on a
    ry_b_local(re, im, p[1]);           // ry(p1) on b
    swap_local(re, im, 2, 3);           // CNOT ctrl=a tgt=b
    ry_b_local(re, im, p[2]);           // ry(p2) on b
    swap_local(re, im, 1, 3);           // CNOT ctrl=b tgt=a
    rz_local(re, im, 1, 0.5f * PI_F);   // rz(pi/2) on a
#pragma unroll
    for (int i = 0; i < 4; ++i) U[i * 4 + tid] = make_float2(re[i], im[i]);
  }
}

// U = Rx(tha) (x) Rx(thb): Rx entry (i,j) = i==j ? c : -i*s  (commuting pair fusion)
__device__ void build_rxrx_U(float tha, float thb, float2* U, int tid) {
  if (tid < 4) {
    float ca = __cosf(0.5f * tha), sa = __sinf(0.5f * tha);
    float cb = __cosf(0.5f * thb), sb = __sinf(0.5f * thb);
    int j = tid, ja = (j >> 1) & 1, jb = j & 1;
#pragma unroll
    for (int i = 0; i < 4; ++i) {
      int ia = (i >> 1) & 1, ib = i & 1;
      float ar = (ia == ja) ? ca : 0.f, ai = (ia == ja) ? 0.f : -sa;
      float br = (ib == jb) ? cb : 0.f, bi = (ib == jb) ? 0.f : -sb;
      U[i * 4 + j] = make_float2(ar * br - ai * bi, ar * bi + ai * br);
    }
  }
}

// spread group index g over N bits, skipping bit positions lo < hi
__device__ __forceinline__ int group_base(int g, int lo, int hi) {
  int lowb  = g & ((1 << lo) - 1);
  int midb  = (g >> lo) & ((1 << (hi - 1 - lo)) - 1);
  int highb = g >> (hi - 1);
  return lowb | (midb << (lo + 1)) | (highb << (hi + 1));
}

// ---------------------------------------------------------------------------
// two-qubit 4x4-complex gate application via V_WMMA_F32_16X16X4_F32
// D(16x16) = A_re(16x4) x [Ur^T|Ui^T] + A_im(16x4) x [-Ui^T|Ur^T]
//   cols 0..3 = Re(out), cols 4..7 = Im(out), rows = 16 state groups.
// The 4-bit WMMA row index M is placed inside one contiguous bit-region of the
// group->amp spread (pigeonhole: a >=4-bit region always exists), so
// amp(M) = base0 + M*stride -> all addresses are linear, no shuffles needed.
// ---------------------------------------------------------------------------
template <int N>
__device__ void conv_apply(float2* st, int pa, int pb, const float2* U, int tid) {
  const int nChunks = 1 << (N - 6);    // (2^(N-2) groups) / 16 rows
  int lane  = tid & 31;
  int wave  = tid >> 5;
  int lo    = pa < pb ? pa : pb;
  int hi    = pa < pb ? pb : pa;
  int maskA = 1 << pa, maskB = 1 << pb;
  int ncol  = lane & 15;
  int kbase = (lane & 16) ? 2 : 0;     // ISA: lanes 0-15 hold K=0,1; lanes 16-31 K=2,3
  int mrow  = (lane & 16) ? 8 : 0;     // D: VGPR r holds row r + 8*(lane>=16)

  // pick a >=4-bit contiguous region of g-space for the row index M
  int rs1 = hi - 1 - lo, rs2 = N - 1 - hi;
  int pstart, adj;
  if (rs2 >= 4)      { pstart = hi - 1; adj = 2; }
  else if (rs1 >= 4) { pstart = lo;     adj = 1; }
  else               { pstart = 0;      adj = 0; }   // then lo >= 4 by pigeonhole
  int stride  = 1 << (pstart + adj);   // amp-index stride between consecutive rows
  int lowmask = (1 << pstart) - 1;

  v2f b1, b2;
#pragma unroll
  for (int t = 0; t < 2; ++t) {
    int k = kbase + t;
    float vr = 0.f, vi = 0.f;
    if (ncol < 4)      { float2 u = U[ncol * 4 + k];       vr = u.x; vi = -u.y; }
    else if (ncol < 8) { float2 u = U[(ncol - 4) * 4 + k]; vr = u.y; vi = u.x;  }
    b1[t] = vr;
    b2[t] = vi;
  }

  int iA     = (kbase & 2) ? maskA : 0;  // m = kbase has a-bit set iff kbase>=2
  // output amp bits for this lane's column: outm bits = ncol bits 0,1 (valid ncol<8)
  int off    = ((ncol & 2) ? maskA : 0) | ((ncol & 1) ? maskB : 0);
  int rowoff = ncol * stride;            // A row M = ncol
  int imoff  = (ncol & 4) ? 1 : 0;       // +1 float -> imag half of float2
  float* stf = (float*)st;

  for (int c = wave; c < nChunks; c += NWAVES) {
    int gbits = ((c >> pstart) << (pstart + 4)) | (c & lowmask);
    int base0 = group_base(gbits, lo, hi);       // uniform across the wave
    int i0 = base0 + rowoff + iA;                // m = kbase
    int i1 = i0 | maskB;                         // m = kbase+1
    float2 s0 = st[i0], s1 = st[i1];
    v2f a1, a2;
    a1[0] = s0.x; a1[1] = s1.x;                  // Re part
    a2[0] = s0.y; a2[1] = s1.y;                  // Im part
    v8f cacc = {0.f, 0.f, 0.f, 0.f, 0.f, 0.f, 0.f, 0.f};
    cacc = __builtin_amdgcn_wmma_f32_16x16x4_f32(false, a1, false, b1, (short)0, cacc, false, false);
    cacc = __builtin_amdgcn_wmma_f32_16x16x4_f32(false, a2, false, b2, (short)0, cacc, false, false);
    if (ncol < 8) {                              // 16 lanes store their own column
      int fbase = ((base0 + mrow * stride + off) << 1) + imoff;  // float-index
      int fstep = stride << 1;
#pragma unroll
      for (int r = 0; r < 8; ++r)
        stf[fbase + r * fstep] = cacc[r];
    }
  }
}

// rx(theta) on one qubit: v0' = c v0 - i s v1 ; v1' = -i s v0 + c v1
template <int N>
__device__ void rx_pass(float2* st, int bitpos, float th, int tid) {
  float c = __cosf(0.5f * th), s = __sinf(0.5f * th);
  const int half = 1 << (N - 1);
  int mask = 1 << bitpos;
  for (int p = tid; p < half; p += NTHR) {
    int i0 = ((p & ~(mask - 1)) << 1) | (p & (mask - 1));
    int i1 = i0 | mask;
    float2 v0 = st[i0], v1 = st[i1];
    st[i0] = make_float2(c * v0.x + s * v1.y, c * v0.y - s * v1.x);
    st[i1] = make_float2(c * v1.x + s * v0.y, c * v1.y - s * v0.x);
  }
}

// fused diagonal layer: all CNOT-Rz-CNOT units = exp(-i th/2 Z(x)Z), one sweep
template <int N>
__device__ void zz_pass(float2* st, const int* masks, const float* th, int tid) {
  const int npairs = 2 * (N / 2);
  const int nAmps  = 1 << N;
  for (int a = tid; a < nAmps; a += NTHR) {
    float phi = 0.f;
#pragma unroll
    for (int k = 0; k < npairs; ++k) {
      float h = 0.5f * th[k];
      phi += (__popc(a & masks[k]) & 1) ? h : -h;   // parity even -> e^{-i th/2}
    }
    float cs = __cosf(phi), sn = __sinf(phi);
    float2 v = st[a];
    st[a] = make_float2(v.x * cs - v.y * sn, v.x * sn + v.y * cs);
  }
}

// ---------------------------------------------------------------------------
// full ansatz block (reps=2), axis w <-> bit N-1-w.
// U construction for gate g+1 is overlapped with application of gate g
// (double-buffered U, one barrier per gate).
// ---------------------------------------------------------------------------
template <int N>
__device__ void ansatz(float2* st, float2* Ubuf /*2x16*/, int* zzm, float* zzt,
                       const float* w, int tid) {
  const int EV = N / 2;          // convs per parity layer (6)
  const int G  = 2 * EV;         // convs per rep (12)
  const int RP = N / 2;          // rx pairs (6)
  int idx = 0;
  for (int rep = 0; rep < 2; ++rep) {
    // ---- conv gates (even pairs then odd pairs), pipelined ----
    build_conv_U(w + idx, Ubuf, tid);
    for (int g = 0; g < G; ++g) {
      __syncthreads();
      if (g + 1 < G) build_conv_U(w + idx + 3 * (g + 1), Ubuf + 16 * ((g + 1) & 1), tid);
      int layer = g / EV;
      int i = 2 * (g % EV) + layer;
      int a = i, b = (i + 1) % N;
      conv_apply<N>(st, N - 1 - a, N - 1 - b, Ubuf + 16 * (g & 1), tid);
    }
    idx += 3 * G;
    __syncthreads();
    // ---- rx layer as Rx(x)Rx pair gates, pipelined ----
    build_rxrx_U(w[idx], w[idx + 1], Ubuf, tid);
    for (int g = 0; g < RP; ++g) {
      __syncthreads();
      if (g + 1 < RP)
        build_rxrx_U(w[idx + 2 * (g + 1)], w[idx + 2 * (g + 1) + 1],
                     Ubuf + 16 * ((g + 1) & 1), tid);
      int q = 2 * g;
      conv_apply<N>(st, N - 1 - q, N - 2 - q, Ubuf + 16 * (g & 1), tid);
    }
    idx += 2 * RP;
    __syncthreads();
    if (N & 1) {                                // leftover qubit N-1 -> bit 0
      rx_pass<N>(st, 0, w[idx++], tid);
      __syncthreads();
    }
    // ---- fused ZZ entangler layer ----
    if (tid == 0) {
      int np = 0;
      for (int layer = 0; layer < 2; ++layer)
        for (int i = layer; i + (layer ? 0 : 1) < N; i += 2) {
          int t = (i + 1) % N;
          zzm[np] = (1 << (N - 1 - i)) | (1 << (N - 1 - t));
          zzt[np] = w[idx + np];
          ++np;
        }
    }
    __syncthreads();
    zz_pass<N>(st, zzm, zzt, tid);
    idx += 2 * (N / 2);
    __syncthreads();
  }
}

// ---------------------------------------------------------------------------
// one workgroup per batch element; state resident in LDS (64KB of 320KB/WGP)
// ---------------------------------------------------------------------------
__global__ __launch_bounds__(NTHR) void pqc_kernel(const float* __restrict__ x,
                                                   const float* __restrict__ wts,
                                                   float* __restrict__ out,
                                                   int wstride) {
  extern __shared__ char smem[];
  float2* st   = (float2*)smem;                                   // 8192 * 8
  float2* Ubuf = (float2*)(smem + 65536);                         // 32 * 8 (double buffer)
  float*  red  = (float*)(smem + 65536 + 256);                    // 16 * 4
  int*    zzm  = (int*)(smem + 65536 + 256 + 64);                 // 16 * 4
  float*  zzt  = (float*)(smem + 65536 + 256 + 64 + 64);          // 16 * 4
  float*  wl   = (float*)(smem + 65536 + 256 + 64 + 64 + 64);     // 368 * 4

  int b = blockIdx.x;
  int tid = threadIdx.x;
  const float* w   = wts + (size_t)b * wstride;
  const float* xin = x + (size_t)b * 4096;

  for (int i = tid; i < 362; i += NTHR) wl[i] = w[i];
  for (int i = tid; i < 4096; i += NTHR) {
    st[i]        = make_float2(xin[i], 0.f);
    st[4096 + i] = make_float2(0.f, 0.f);                         // 13th-qubit expansion
  }
  __syncthreads();

  ansatz<12>(st, Ubuf, zzm, zzt, wl, tid);          // block 1 (params 0..119)
  ansatz<13>(st, Ubuf, zzm, zzt, wl + 120, tid);    // block 2 (params 120..241)

  // norm over first 4096 amps, then s2n = s2s / (norm + 1e-8)
  if (tid == 0) red[0] = 0.f;
  __syncthreads();
  float part = 0.f;
  for (int i = tid; i < 4096; i += NTHR) {
    float2 v = st[i];
    part = __fmaf_rn(v.x, v.x, part);
    part = __fmaf_rn(v.y, v.y, part);
  }
  atomicAdd(&red[0], part);
  __syncthreads();
  float inv = 1.f / (__fsqrt_rn(red[0]) + 1e-8f);
  for (int i = tid; i < 4096; i += NTHR) {
    st[i].x *= inv;
    st[i].y *= inv;
  }
  __syncthreads();

  ansatz<12>(st, Ubuf, zzm, zzt, wl + 242, tid);    // block 3 (params 242..361)

  // expvals: es[q] = sum_p |amp|^2 * (bit(11-q)==0 ? +1 : -1)
  if (tid < 12) red[tid] = 0.f;
  __syncthreads();
  float acc[12];
#pragma unroll
  for (int q = 0; q < 12; ++q) acc[q] = 0.f;
  for (int i = tid; i < 4096; i += NTHR) {
    float2 v = st[i];
    float p = v.x * v.x + v.y * v.y;
#pragma unroll
    for (int q = 0; q < 12; ++q)
      acc[q] += ((i >> (11 - q)) & 1) ? -p : p;
  }
#pragma unroll
  for (int q = 0; q < 12; ++q) atomicAdd(&red[q], acc[q]);
  __syncthreads();
  if (tid < 12) out[(size_t)b * 12 + tid] = red[tid];
}

extern "C" void kernel_launch(void* const* d_in, const int* in_sizes, int n_in,
                              void* d_out, int out_size, void* d_ws, size_t ws_size,
                              hipStream_t stream) {
  const float* x = (const float*)d_in[0];
  const float* w = (const float*)d_in[1];
  float* out = (float*)d_out;
  int batch   = in_sizes[0] / 4096;
  int wstride = in_sizes[1] / batch;               // 362
  size_t lds  = 65536 + 256 + 64 + 64 + 64 + 368 * 4;
  (void)hipFuncSetAttribute((const void*)pqc_kernel,
                            hipFuncAttributeMaxDynamicSharedMemorySize, (int)lds);
  pqc_kernel<<<batch, NTHR, lds, stream>>>(x, w, out, wstride);
  (void)d_ws; (void)ws_size; (void)n_in; (void)out_size;
}